// SpaceEffect_42176578847123
// MI455X (gfx1250) — compile-verified
//
#include <hip/hip_runtime.h>
#include <hip/hip_bf16.h>

typedef __attribute__((ext_vector_type(2))) float v2f;
typedef __attribute__((ext_vector_type(8))) float v8f;

#define T_DIM 12
#define N_DIM 1000
#define D_DIM 128
#define N_TILES 63                      // ceil(1000/16)
#define SCALE 0.08838834764831845f     // 1/sqrt(128)

// ---------------------------------------------------------------------------
// K1: scores[t] = mask(adj, (X[t] @ X[t]^T) * scale)
// One wave per 16x16 tile. Out-of-range tile rows are CLAMPED (not zeroed):
// K is always fully valid, so clamped-lane garbage only reaches output
// elements that are masked at store time. No divergence in the inner loop.
// ---------------------------------------------------------------------------
__global__ __launch_bounds__(32)
void gat_scores_kernel(const float* __restrict__ X, const int* __restrict__ adj,
                       float* __restrict__ scores) {
    const int rt   = blockIdx.x;          // query row tile
    const int ct   = blockIdx.y;          // key   col tile
    const int t    = blockIdx.z;
    const int lane = threadIdx.x;
    const int lrow = lane & 15;
    const int half = lane >> 4;

    const float* Xt = X + (size_t)t * N_DIM * D_DIM;
    const int arow = min(rt * 16 + lrow, N_DIM - 1);
    const int brow = min(ct * 16 + lrow, N_DIM - 1);
    const float* Ap = Xt + (size_t)arow * D_DIM + 2 * half;
    const float* Bp = Xt + (size_t)brow * D_DIM + 2 * half;

    v8f acc = {};
#pragma unroll
    for (int k0 = 0; k0 < D_DIM; k0 += 4) {
        const v2f a = *(const v2f*)(Ap + k0);
        const v2f b = *(const v2f*)(Bp + k0);
        acc = __builtin_amdgcn_wmma_f32_16x16x4_f32(false, a, false, b,
                                                    (short)0, acc, false, false);
    }

    float* St = scores + (size_t)t * N_DIM * N_DIM;
    const int col = ct * 16 + lrow;       // C/D: N = lane%16
    if (col < N_DIM) {
#pragma unroll
        for (int g = 0; g < 8; ++g) {
            const int row = rt * 16 + g + 8 * half;   // C/D: M = vgpr + 8*(lane/16)
            if (row < N_DIM) {
                float v = acc[g] * SCALE;
                v = (adj[row * N_DIM + col] > 0) ? v : -1.0e12f;
                St[(size_t)row * N_DIM + col] = v;
            }
        }
    }
}

// ---------------------------------------------------------------------------
// K2: in-place row softmax over [T*N, N]
// ---------------------------------------------------------------------------
__global__ __launch_bounds__(256)
void softmax_kernel(float* __restrict__ scores) {
    __shared__ float sm[256];
    const int tid = threadIdx.x;
    float* S = scores + (size_t)blockIdx.x * N_DIM;

    float m = -3.0e38f;
    for (int i = tid; i < N_DIM; i += 256) m = fmaxf(m, S[i]);
    sm[tid] = m;
    __syncthreads();
    for (int s = 128; s > 0; s >>= 1) {
        if (tid < s) sm[tid] = fmaxf(sm[tid], sm[tid + s]);
        __syncthreads();
    }
    m = sm[0];
    __syncthreads();

    float sum = 0.0f;
    for (int i = tid; i < N_DIM; i += 256) {
        const float e = expf(S[i] - m);
        S[i] = e;
        sum += e;
    }
    sm[tid] = sum;
    __syncthreads();
    for (int s = 128; s > 0; s >>= 1) {
        if (tid < s) sm[tid] += sm[tid + s];
        __syncthreads();
    }
    const float inv = 1.0f / sm[0];
    for (int i = tid; i < N_DIM; i += 256) S[i] *= inv;
}

// ---------------------------------------------------------------------------
// K3: node_features[t] = attn[t] (1000x1000) @ X[t] (1000x128)
// K = 1000 (divisible by 4) is fully valid; only A's row index needs
// clamping, with store-time masking of padded output rows.
// ---------------------------------------------------------------------------
__global__ __launch_bounds__(32)
void gat_aggregate_kernel(const float* __restrict__ attn,
                          const float* __restrict__ X,
                          float* __restrict__ nf) {
    const int rt   = blockIdx.x;          // row tile over N
    const int dt   = blockIdx.y;          // col tile over D (8 tiles)
    const int t    = blockIdx.z;
    const int lane = threadIdx.x;
    const int lrow = lane & 15;
    const int half = lane >> 4;

    const float* At = attn + (size_t)t * N_DIM * N_DIM;
    const float* Xt = X + (size_t)t * N_DIM * D_DIM;
    const int arow = min(rt * 16 + lrow, N_DIM - 1);
    const int dcol = dt * 16 + lrow;

    const float* Ap = At + (size_t)arow * N_DIM + 2 * half;
    const float* Bp = Xt + (size_t)(2 * half) * D_DIM + dcol;

    v8f acc = {};
#pragma unroll 10
    for (int k0 = 0; k0 < N_DIM; k0 += 4) {
        const v2f a = *(const v2f*)(Ap + k0);
        v2f b;
        b.x = Bp[(size_t)k0 * D_DIM];             // B vgpr0: K = k0+2*half
        b.y = Bp[(size_t)(k0 + 1) * D_DIM];       // B vgpr1: K = k0+2*half+1
        acc = __builtin_amdgcn_wmma_f32_16x16x4_f32(false, a, false, b,
                                                    (short)0, acc, false, false);
    }

    float* Nt = nf + (size_t)t * N_DIM * D_DIM;
#pragma unroll
    for (int g = 0; g < 8; ++g) {
        const int row = rt * 16 + g + 8 * half;
        if (row < N_DIM) Nt[(size_t)row * D_DIM + dcol] = acc[g];
    }
}

// ---------------------------------------------------------------------------
// K4: per-node temporal Gram: W[j] = sigmoid( (sf[j] @ sf[j]^T) * scale )
// sf[j][t][d] = nf[t][j][d]. One wave per node; time index clamped to 11,
// padded rows/cols masked at store. A-frag == B-frag (symmetric Gram).
// ---------------------------------------------------------------------------
__global__ __launch_bounds__(32)
void temporal_kernel(const float* __restrict__ nf, float* __restrict__ W) {
    const int j    = blockIdx.x;
    const int lane = threadIdx.x;
    const int lrow = lane & 15;
    const int half = lane >> 4;
    const int trow = min(lrow, T_DIM - 1);

    const float* base = nf + (size_t)trow * N_DIM * D_DIM + (size_t)j * D_DIM
                           + 2 * half;

    v8f acc = {};
#pragma unroll
    for (int k0 = 0; k0 < D_DIM; k0 += 4) {
        const v2f a = *(const v2f*)(base + k0);
        acc = __builtin_amdgcn_wmma_f32_16x16x4_f32(false, a, false, a,
                                                    (short)0, acc, false, false);
    }

    float* Wj = W + (size_t)j * T_DIM * T_DIM;
    if (lrow < T_DIM) {
#pragma unroll
        for (int g = 0; g < 8; ++g) {
            const int t1 = g + 8 * half;
            if (t1 < T_DIM) {
                const float v = acc[g] * SCALE;
                Wj[t1 * T_DIM + lrow] = 1.0f / (1.0f + expf(-v));
            }
        }
    }
}

// ---------------------------------------------------------------------------
// K5: out[(t2*N + j)*12000 + t1*1000 + i] = W[j][t1][t2]   (broadcast over i)
// One block per output row; coalesced float4 (global_store_b128) writes.
// This is the 576 MB HBM-bound stage: ~25 us floor at 23.3 TB/s.
// ---------------------------------------------------------------------------
__global__ __launch_bounds__(256)
void broadcast_kernel(const float* __restrict__ W, float* __restrict__ out) {
    const int rowIdx = blockIdx.x;        // 0 .. T*N-1
    const int t2 = rowIdx / N_DIM;
    const int j  = rowIdx - t2 * N_DIM;

    __shared__ float w[T_DIM];
    if (threadIdx.x < T_DIM)
        w[threadIdx.x] = W[(size_t)j * T_DIM * T_DIM + threadIdx.x * T_DIM + t2];
    __syncthreads();

    float4* o = (float4*)(out + (size_t)rowIdx * (T_DIM * N_DIM));
    const int nvec = T_DIM * N_DIM / 4;   // 3000 float4 per row
    for (int i = threadIdx.x; i < nvec; i += 256) {
        const float val = w[i / 250];     // i*4/1000
        o[i] = make_float4(val, val, val, val);
    }
}

// ---------------------------------------------------------------------------
extern "C" void kernel_launch(void* const* d_in, const int* in_sizes, int n_in,
                              void* d_out, int out_size, void* d_ws, size_t ws_size,
                              hipStream_t stream) {
    const float* X   = (const float*)d_in[0];   // [T,N,D] f32
    const int*   adj = (const int*)d_in[1];     // [N,N] i32
    float* out = (float*)d_out;                 // [T*N, T*N] f32

    // Workspace partition (~54.7 MB total)
    char* ws = (char*)d_ws;
    float* scores = (float*)ws;                                        // T*N*N  = 48,000,000 B
    float* nf     = (float*)(ws + (size_t)T_DIM * N_DIM * N_DIM * 4);  // T*N*D  =  6,144,000 B
    float* W      = (float*)(ws + (size_t)T_DIM * N_DIM * N_DIM * 4
                                + (size_t)T_DIM * N_DIM * D_DIM * 4);  // N*T*T  =    576,000 B

    dim3 g1(N_TILES, N_TILES, T_DIM);
    gat_scores_kernel<<<g1, 32, 0, stream>>>(X, adj, scores);

    softmax_kernel<<<T_DIM * N_DIM, 256, 0, stream>>>(scores);

    dim3 g3(N_TILES, D_DIM / 16, T_DIM);
    gat_aggregate_kernel<<<g3, 32, 0, stream>>>(scores, X, nf);

    temporal_kernel<<<N_DIM, 32, 0, stream>>>(nf, W);

    broadcast_kernel<<<T_DIM * N_DIM, 256, 0, stream>>>(W, out);
}